// LIF_21260088115614
// MI455X (gfx1250) — compile-verified
//
#include <hip/hip_runtime.h>
#include <stdint.h>

typedef float v2f __attribute__((ext_vector_type(2)));
typedef float v8f __attribute__((ext_vector_type(8)));

#define NSTEPS 100
#define CIN    256
#define OUP    256
#define NROWS  16           // WMMA M tile = samples per workgroup
#define LDSTR  260          // padded LDS row stride (floats) -> conflict-free ds_load_b64
#define LDSBUF (NROWS * LDSTR)

#define ALPHA_F 0.81873075f // exp(-1e-3/5e-3)
#define BETA_F  0.90483743f // exp(-1e-3/1e-2)

// Async copy of this thread's 64-byte chunk of the A tile directly into LDS.
// CDNA5 GLOBAL_LOAD_ASYNC_TO_LDS: INST_OFFSET is added to BOTH the global and
// the LDS address, so one (ldsaddr, voff) pair covers all four b128 beats.
__device__ __forceinline__ void async_tile_load(unsigned ldsaddr, unsigned voff,
                                                const float* __restrict__ base)
{
  asm volatile(
      "global_load_async_to_lds_b128 %0, %1, %2 offset:0\n\t"
      "global_load_async_to_lds_b128 %0, %1, %2 offset:16\n\t"
      "global_load_async_to_lds_b128 %0, %1, %2 offset:32\n\t"
      "global_load_async_to_lds_b128 %0, %1, %2 offset:48"
      :
      : "v"(ldsaddr), "v"(voff), "s"(base)
      : "memory");
}

__device__ __forceinline__ void wait_async0()
{
#if __has_builtin(__builtin_amdgcn_s_wait_asynccnt)
  __builtin_amdgcn_s_wait_asynccnt(0);
#else
  asm volatile("s_wait_asynccnt 0" ::: "memory");
#endif
}

// One LIF scan step + FIR smoothing for time s, on a 16x16 register tile.
// v8f element i maps to output row (8*hi + i), col (lane&15)  [WMMA C layout].
__device__ __forceinline__ void lif_emit(
    int s,
    const v8f& rm2, const v8f& rm1, const v8f& r0,
    const v8f& rp1, const v8f& rp2,
    float e0, float e1, float e2, float e3, float e4,
    v8f& syn, v8f& mem,
    float* __restrict__ mem_base, float* __restrict__ spk_base)
{
#pragma unroll
  for (int i = 0; i < 8; ++i) {
    float m = mem[i];
    float o = (m - 1.0f > 0.0f) ? 1.0f : 0.0f;            // spike (records use OLD mem)
    float h = e0 * rm2[i] + e1 * rm1[i] + e2 * r0[i] + e3 * rp1[i] + e4 * rp2[i];
    size_t off = (size_t)i * (NSTEPS * OUP) + (size_t)s * OUP;
    __builtin_nontemporal_store(m, &mem_base[off]);       // write-once records: keep out of L2
    __builtin_nontemporal_store(o, &spk_base[off]);
    float nm = (BETA_F * m + syn[i]) * (1.0f - o);        // uses old syn
    syn[i] = ALPHA_F * syn[i] + h;
    mem[i] = nm;
  }
}

__global__ __launch_bounds__(256) void lif_fused_kernel(
    const float* __restrict__ inp,   // (nsamp, 100, 256)
    const float* __restrict__ w,     // (256, 256)
    const float* __restrict__ eps,   // (5)
    float* __restrict__ out,         // mem_rec | spk_rec, each (nsamp,100,256)
    int nsamp)
{
  __shared__ float lds[2 * LDSBUF];  // double-buffered A tile (16 x 256, padded)

  const int tid  = threadIdx.x;      // 0..255 (8 waves, wave32)
  const int lane = tid & 31;
  const int wave = tid >> 5;
  const int hi   = lane >> 4;        // half-wave select
  const int lo   = lane & 15;

  const int n0 = blockIdx.x * NROWS;              // sample tile base
  const int d0 = blockIdx.y * 128 + wave * 16;    // output column tile base

  // ---- Resident B fragments: w[0:256, d0:d0+16] as 64 K-groups of 16x16x4 ----
  // B layout (f32 4x16): lane<16 holds (K=4k+0,4k+1) of col=lane; lane>=16 holds (K=4k+2,4k+3).
  v2f Bf[64];
  {
    const float* wp = w + (size_t)(2 * hi) * OUP + (d0 + lo);
#pragma unroll
    for (int k = 0; k < 64; ++k) {
      Bf[k].x = wp[(size_t)(4 * k) * OUP];
      Bf[k].y = wp[(size_t)(4 * k + 1) * OUP];
    }
  }

  const float e0 = eps[0], e1 = eps[1], e2 = eps[2], e3 = eps[3], e4 = eps[4];

  float* mem_base = out + (size_t)(n0 + 8 * hi) * (NSTEPS * OUP) + d0 + lo;
  float* spk_base = mem_base + (size_t)nsamp * NSTEPS * OUP;

  // ---- async staging geometry: 256 threads cover 16 rows x 256 floats ----
  const int lrow = tid >> 4;          // 0..15
  const int lcol = (tid & 15) * 16;   // 0..240 step 16 (64B chunk per thread)
  // LDS byte addresses of this thread's chunk in each buffer (flat ptr low 32 bits = LDS offset)
  const unsigned lds0 = (unsigned)(uintptr_t)&lds[lrow * LDSTR + lcol];
  const unsigned lds1 = lds0 + (unsigned)(LDSBUF * 4);
  // 32-bit global byte offset of this thread's chunk at t=0
  const unsigned voff0 = (unsigned)(((n0 + lrow) * NSTEPS) * CIN + lcol) * 4u;

  // preload t = 0 into buffer 0
  async_tile_load(lds0, voff0, inp);
  wait_async0();
  __syncthreads();

  const v8f vzero = {};
  v8f ring[5];
#pragma unroll
  for (int i = 0; i < 5; ++i) ring[i] = vzero;   // zero-pad left boundary for FIR
  v8f syn = {}, mem = {};

  const float* Abase = &lds[lo * LDSTR + 2 * hi]; // A frag: row=lo, K pair 2*hi within group

  for (int tb = 0; tb < NSTEPS / 5; ++tb) {
#pragma unroll
    for (int p = 0; p < 5; ++p) {                 // ring slot p is compile-time
      const int t = tb * 5 + p;
      const int cur = t & 1;

      // kick off next-step tile copy straight into the other LDS buffer
      if (t + 1 < NSTEPS) {
        async_tile_load(cur ? lds0 : lds1,
                        voff0 + (unsigned)(t + 1) * (CIN * 4u), inp);
      }

      // GEMM: (16x256) @ (256x16) = 64 x v_wmma_f32_16x16x4_f32, 2 acc chains
      v8f acc0 = {}, acc1 = {};
      const float* A = Abase + cur * LDSBUF;
#pragma unroll
      for (int k = 0; k < 64; k += 2) {
        v2f a0 = *(const v2f*)(A + 4 * k);
        v2f a1 = *(const v2f*)(A + 4 * k + 4);
        acc0 = __builtin_amdgcn_wmma_f32_16x16x4_f32(
            false, a0, false, Bf[k], (short)0, acc0, false, false);
        acc1 = __builtin_amdgcn_wmma_f32_16x16x4_f32(
            false, a1, false, Bf[k + 1], (short)0, acc1, false, false);
      }
      ring[p] = acc0 + acc1;

      // retire time s = t-2 (raw[t-4..t] available; zero-init ring pads left edge)
      if (t >= 2) {
        lif_emit(t - 2,
                 ring[(p + 1) % 5], ring[(p + 2) % 5], ring[(p + 3) % 5],
                 ring[(p + 4) % 5], ring[p],
                 e0, e1, e2, e3, e4, syn, mem, mem_base, spk_base);
      }

      // make this wave's async writes complete, then publish to all waves
      wait_async0();
      __syncthreads();
    }
  }

  // right-edge flush: s=98 uses raw[96..99] (slots 1,2,3,4), s=99 uses raw[97..99]
  lif_emit(98, ring[1], ring[2], ring[3], ring[4], vzero,
           e0, e1, e2, e3, e4, syn, mem, mem_base, spk_base);
  lif_emit(99, ring[2], ring[3], ring[4], vzero, vzero,
           e0, e1, e2, e3, e4, syn, mem, mem_base, spk_base);
}

extern "C" void kernel_launch(void* const* d_in, const int* in_sizes, int n_in,
                              void* d_out, int out_size, void* d_ws, size_t ws_size,
                              hipStream_t stream) {
  const float* inp = (const float*)d_in[0];
  const float* w   = (const float*)d_in[1];
  const float* eps = (const float*)d_in[2];
  float* out = (float*)d_out;
  const int nsamp = in_sizes[0] / (NSTEPS * CIN);   // 2048
  dim3 grid(nsamp / NROWS, OUP / 128);              // (128, 2)
  lif_fused_kernel<<<grid, dim3(256), 0, stream>>>(inp, w, eps, out, nsamp);
}